// PointnetSAModuleMSG_Torch_30511447670986
// MI455X (gfx1250) — compile-verified
//
#include <hip/hip_runtime.h>
#include <hip/hip_bf16.h>
#include <math.h>

typedef __attribute__((ext_vector_type(2))) float v2f;
typedef __attribute__((ext_vector_type(8))) float v8f;

#define BATCH 8
#define NPTS  16384
#define NQ    1024
#define EPSBN 1e-5f

// ---------------------------------------------------------------- FPS
// One block per batch. dist[] lives in global ws (hot in L2), argmax via LDS tree.
__global__ void fps_kernel(const float* __restrict__ xyz, float* __restrict__ newxyz,
                           float* __restrict__ dist)
{
    __shared__ float sval[1024];
    __shared__ int   sidx[1024];
    __shared__ int   s_far;
    const int b   = blockIdx.x;
    const int tid = threadIdx.x;
    const float* x = xyz + (size_t)b * NPTS * 3;
    float* d = dist + (size_t)b * NPTS;
    for (int n = tid; n < NPTS; n += blockDim.x) d[n] = 3.4e38f;
    if (tid == 0) s_far = 0;
    __syncthreads();
    for (int k = 0; k < NQ; ++k) {
        const int far = s_far;
        const float cx = x[far*3+0], cy = x[far*3+1], cz = x[far*3+2];
        if (tid == 0) {
            newxyz[((size_t)b*NQ + k)*3 + 0] = cx;
            newxyz[((size_t)b*NQ + k)*3 + 1] = cy;
            newxyz[((size_t)b*NQ + k)*3 + 2] = cz;
        }
        float bestv = -1.0f; int besti = 0;
        for (int n = tid; n < NPTS; n += blockDim.x) {
            float dx = x[n*3+0]-cx, dy = x[n*3+1]-cy, dz = x[n*3+2]-cz;
            float dd = dx*dx + dy*dy + dz*dz;
            dd = fminf(d[n], dd);
            d[n] = dd;
            if (dd > bestv || (dd == bestv && n < besti)) { bestv = dd; besti = n; }
        }
        sval[tid] = bestv; sidx[tid] = besti;
        __syncthreads();
        for (int s = blockDim.x >> 1; s > 0; s >>= 1) {
            if (tid < s) {
                float v2 = sval[tid+s]; int i2 = sidx[tid+s];
                if (v2 > sval[tid] || (v2 == sval[tid] && i2 < sidx[tid])) {
                    sval[tid] = v2; sidx[tid] = i2;
                }
            }
            __syncthreads();
        }
        if (tid == 0) s_far = sidx[0];
        __syncthreads();
    }
}

// ---------------------------------------------------------------- Ball query
// One thread per (b,q); block covers 256 queries of ONE batch (1024 % 256 == 0),
// so the block stages xyz in LDS tiles and scans out of LDS.
#define BQ_TILE 1024
__global__ void ball_query_kernel(const float* __restrict__ xyz,
                                  const float* __restrict__ newxyz,
                                  int* __restrict__ idx, float r2, int K)
{
    __shared__ float sx[BQ_TILE], sy[BQ_TILE], sz[BQ_TILE];
    const int t = blockIdx.x * blockDim.x + threadIdx.x;   // 0 .. BATCH*NQ-1
    const int b = t / NQ;
    const float* x = xyz + (size_t)b * NPTS * 3;
    const float cx = newxyz[t*3+0], cy = newxyz[t*3+1], cz = newxyz[t*3+2];
    float bd[128]; int bi[128]; int cnt = 0;
    for (int base = 0; base < NPTS; base += BQ_TILE) {
        __syncthreads();
        for (int i = threadIdx.x; i < BQ_TILE; i += blockDim.x) {
            sx[i] = x[(base+i)*3+0];
            sy[i] = x[(base+i)*3+1];
            sz[i] = x[(base+i)*3+2];
        }
        __syncthreads();
        for (int j = 0; j < BQ_TILE; ++j) {
            const float dx = sx[j]-cx, dy = sy[j]-cy, dz = sz[j]-cz;
            const float d2 = dx*dx + dy*dy + dz*dz;
            if (d2 > r2) continue;
            const int n = base + j;
            if (cnt < K) {
                int p = cnt++;
                while (p > 0 && bd[p-1] > d2) { bd[p] = bd[p-1]; bi[p] = bi[p-1]; --p; }
                bd[p] = d2; bi[p] = n;
            } else if (d2 < bd[K-1]) {
                int p = K - 1;
                while (p > 0 && bd[p-1] > d2) { bd[p] = bd[p-1]; bi[p] = bi[p-1]; --p; }
                bd[p] = d2; bi[p] = n;
            }
        }
    }
    int* o = idx + (size_t)t * K;
    for (int k = 0; k < K; ++k) o[k] = (k < cnt) ? bi[k] : -1;
}

// ---------------------------------------------------------------- Gather
// Build channel-planar G[68][M]: 64 feature ch + 3 relative xyz + 1 zero pad.
__global__ void gather_kernel(const float* __restrict__ xyz,
                              const float* __restrict__ feat,
                              const float* __restrict__ newxyz,
                              const int* __restrict__ idx,
                              float* __restrict__ G, int K, size_t M)
{
    const size_t m = (size_t)blockIdx.x * blockDim.x + threadIdx.x;
    if (m >= M) return;
    const size_t bq = m / (size_t)K;          // b*NQ + q
    const int b = (int)(bq / NQ);
    const int id = idx[m];
    if (id < 0) {
        for (int c = 0; c < 68; ++c) G[(size_t)c * M + m] = 0.0f;
        return;
    }
    const float* f = feat + (size_t)b * 64 * NPTS;
    for (int c = 0; c < 64; ++c) G[(size_t)c * M + m] = f[(size_t)c * NPTS + id];
    const float* p  = xyz + ((size_t)b * NPTS + id) * 3;
    const float* c0 = newxyz + bq * 3;
    G[(size_t)64 * M + m] = p[0] - c0[0];
    G[(size_t)65 * M + m] = p[1] - c0[1];
    G[(size_t)66 * M + m] = p[2] - c0[2];
    G[(size_t)67 * M + m] = 0.0f;
}

// ---------------------------------------------------------------- W padding (Cin 67 -> 68)
__global__ void pad_w_kernel(const float* __restrict__ W, float* __restrict__ Wp,
                             int Cout, int Cin, int Cp)
{
    const int t = blockIdx.x * blockDim.x + threadIdx.x;
    if (t >= Cout * Cp) return;
    const int o = t / Cp, c = t % Cp;
    Wp[t] = (c < Cin) ? W[o * Cin + c] : 0.0f;
}

// ---------------------------------------------------------------- WMMA GEMM (f32)
// Y[Cout x M] = W[Cout x Cin] * G[Cin x M]. One wave per 16(rows) x 64(cols)
// output tile: 4 accumulators share one A operand per K-step
// -> 4x V_WMMA_F32_16X16X4_F32 per A fetch.
__global__ void gemm_wmma_kernel(const float* __restrict__ Wm,
                                 const float* __restrict__ G,
                                 float* __restrict__ Y, int Cin, size_t M)
{
    const int lane = threadIdx.x;          // 0..31, full wave active (EXEC all-1s)
    const int half = lane >> 4;            // 0: K pair {0,1}, 1: K pair {2,3}
    const int l    = lane & 15;
    const size_t mt = (size_t)blockIdx.x * 64;
    const int    ot = blockIdx.y * 16;
    const int    kb = half * 2;
    v8f a0 = {0.f,0.f,0.f,0.f,0.f,0.f,0.f,0.f};
    v8f a1 = a0, a2 = a0, a3 = a0;
    const float* wrow = Wm + (size_t)(ot + l) * Cin + kb;  // A: row = M-index l
    const float* g0   = G + mt + l;                        // B: col = N-index l
    for (int c = 0; c < Cin; c += 4) {
        v2f a;
        a.x = wrow[c];
        a.y = wrow[c + 1];
        const float* gr0 = g0 + (size_t)(c + kb)     * M;
        const float* gr1 = g0 + (size_t)(c + kb + 1) * M;
        v2f b0, b1, b2, b3;
        b0.x = gr0[0];  b0.y = gr1[0];
        b1.x = gr0[16]; b1.y = gr1[16];
        b2.x = gr0[32]; b2.y = gr1[32];
        b3.x = gr0[48]; b3.y = gr1[48];
        a0 = __builtin_amdgcn_wmma_f32_16x16x4_f32(false, a, false, b0, (short)0, a0, false, false);
        a1 = __builtin_amdgcn_wmma_f32_16x16x4_f32(false, a, false, b1, (short)0, a1, false, false);
        a2 = __builtin_amdgcn_wmma_f32_16x16x4_f32(false, a, false, b2, (short)0, a2, false, false);
        a3 = __builtin_amdgcn_wmma_f32_16x16x4_f32(false, a, false, b3, (short)0, a3, false, false);
    }
    float* yo = Y + mt + l;
#pragma unroll
    for (int r = 0; r < 8; ++r) {
        const size_t row = (size_t)(ot + r + 8*half) * M;  // C/D layout: M = r + 8*(lane/16)
        yo[row +  0] = a0[r];
        yo[row + 16] = a1[r];
        yo[row + 32] = a2[r];
        yo[row + 48] = a3[r];
    }
}

// ---------------------------------------------------------------- BN helpers
__global__ void zero_kernel(float* __restrict__ p, int n)
{
    const int i = blockIdx.x * blockDim.x + threadIdx.x;
    if (i < n) p[i] = 0.0f;
}

__global__ void bn_stats_kernel(const float* __restrict__ Y, float* __restrict__ stats,
                                int Cout, size_t M)
{
    __shared__ float ss[256], sq[256];
    const int c = blockIdx.y;
    const float* y = Y + (size_t)c * M;
    float s = 0.f, q = 0.f;
    for (size_t i = (size_t)blockIdx.x * blockDim.x + threadIdx.x; i < M;
         i += (size_t)gridDim.x * blockDim.x) {
        const float v = y[i]; s += v; q += v * v;
    }
    ss[threadIdx.x] = s; sq[threadIdx.x] = q;
    __syncthreads();
    for (int st = 128; st > 0; st >>= 1) {
        if (threadIdx.x < st) {
            ss[threadIdx.x] += ss[threadIdx.x + st];
            sq[threadIdx.x] += sq[threadIdx.x + st];
        }
        __syncthreads();
    }
    if (threadIdx.x == 0) {
        atomicAdd(&stats[c], ss[0]);
        atomicAdd(&stats[Cout + c], sq[0]);
    }
}

__global__ void bn_relu_kernel(float* __restrict__ Y, const float* __restrict__ stats,
                               const float* __restrict__ gamma, const float* __restrict__ beta,
                               int Cout, size_t M)
{
    const int c = blockIdx.y;
    const float mean = stats[c] / (float)M;
    const float var  = stats[Cout + c] / (float)M - mean * mean;
    const float sc = gamma[c] * rsqrtf(var + EPSBN);
    const float bs = beta[c] - mean * sc;
    float* y = Y + (size_t)c * M;
    for (size_t i = (size_t)blockIdx.x * blockDim.x + threadIdx.x; i < M;
         i += (size_t)gridDim.x * blockDim.x) {
        const float v = y[i] * sc + bs;
        y[i] = v > 0.f ? v : 0.f;
    }
}

// ---------------------------------------------------------------- Max-pool over K
__global__ void maxpool_kernel(const float* __restrict__ Y, float* __restrict__ out,
                               int K, int coff)
{
    const int q = blockIdx.x * blockDim.x + threadIdx.x;
    const int o = blockIdx.y;
    const int b = blockIdx.z;
    if (q >= NQ) return;
    const size_t M = (size_t)BATCH * NQ * K;
    const float* y = Y + (size_t)o * M + ((size_t)(b * NQ + q)) * K;
    float mx = -3.4e38f;
    for (int k = 0; k < K; ++k) mx = fmaxf(mx, y[k]);
    out[((size_t)b * 320 + coff + o) * NQ + q] = mx;
}

// ---------------------------------------------------------------- launch
extern "C" void kernel_launch(void* const* d_in, const int* in_sizes, int n_in,
                              void* d_out, int out_size, void* d_ws, size_t ws_size,
                              hipStream_t stream) {
    (void)in_sizes; (void)n_in; (void)out_size; (void)ws_size;
    const float* xyz  = (const float*)d_in[0];
    const float* feat = (const float*)d_in[1];
    // params flattened depth-first: branch -> layer -> (W, gamma, beta)
    const float *Wt[3][3], *Gm[3][3], *Bt[3][3];
    int pi = 2;
    for (int br = 0; br < 3; ++br)
        for (int l = 0; l < 3; ++l) {
            Wt[br][l] = (const float*)d_in[pi++];
            Gm[br][l] = (const float*)d_in[pi++];
            Bt[br][l] = (const float*)d_in[pi++];
        }

    static const int   cins [3][3] = {{68,32,32},{68,64,64},{68,64,96}}; // padded Cin for layer 0
    static const int   rawci[3][3] = {{67,32,32},{67,64,64},{67,64,96}};
    static const int   couts[3][3] = {{32,32,64},{64,64,128},{64,96,128}};
    static const int   Ks[3]    = {16, 32, 128};
    static const float r2s[3]   = {0.01f, 0.04f, 0.16f};
    static const int   coffs[3] = {0, 64, 192};

    // workspace carve-out
    char* w = (char*)d_ws;
    size_t off = 0;
    auto carve = [&](size_t bytes) { size_t r = off; off = (off + bytes + 255) & ~(size_t)255; return r; };
    float* dist  = (float*)(w + carve((size_t)BATCH * NPTS * 4));
    int*   idx   = (int*)  (w + carve((size_t)BATCH * NQ * 128 * 4));
    float* Wpad  = (float*)(w + carve((size_t)128 * 68 * 4));
    float* stats = (float*)(w + carve((size_t)256 * 4));
    const size_t Mmax = (size_t)BATCH * NQ * 128;
    float* P0 = (float*)(w + carve(Mmax * 128 * 4));
    float* P1 = (float*)(w + carve(Mmax * 128 * 4));

    float* out_xyz  = (float*)d_out;                    // [8,1024,3]
    float* out_feat = out_xyz + (size_t)BATCH * NQ * 3; // [8,320,1024]

    // 1) FPS -> new_xyz directly into d_out
    fps_kernel<<<BATCH, 1024, 0, stream>>>(xyz, out_xyz, dist);

    for (int br = 0; br < 3; ++br) {
        const int K = Ks[br];
        const size_t M = (size_t)BATCH * NQ * K;

        // 2) ball query + gather into P0 (68 channel-planes)
        ball_query_kernel<<<(BATCH * NQ + 255) / 256, 256, 0, stream>>>(
            xyz, out_xyz, idx, r2s[br], K);
        gather_kernel<<<(unsigned)((M + 255) / 256), 256, 0, stream>>>(
            xyz, feat, out_xyz, idx, P0, K, M);

        // 3) MLP: GEMM (WMMA f32) -> BN(train stats) -> ReLU, ping-pong P0/P1
        float* ping = P0;
        float* pong = P1;
        const int rblocks = (int)((M + 255) / 256 > 1024 ? 1024 : (M + 255) / 256);
        for (int l = 0; l < 3; ++l) {
            const int Cin  = cins[br][l];
            const int Cout = couts[br][l];
            const float* Wuse = Wt[br][l];
            if (l == 0) {   // zero-pad Cin 67 -> 68
                pad_w_kernel<<<(Cout * Cin + 255) / 256, 256, 0, stream>>>(
                    Wt[br][l], Wpad, Cout, rawci[br][l], Cin);
                Wuse = Wpad;
            }
            dim3 ggrid((unsigned)(M / 64), (unsigned)(Cout / 16));
            gemm_wmma_kernel<<<ggrid, 32, 0, stream>>>(Wuse, ping, pong, Cin, M);

            zero_kernel<<<1, 256, 0, stream>>>(stats, 2 * Cout);
            dim3 sgrid((unsigned)rblocks, (unsigned)Cout);
            bn_stats_kernel<<<sgrid, 256, 0, stream>>>(pong, stats, Cout, M);
            bn_relu_kernel<<<sgrid, 256, 0, stream>>>(pong, stats, Gm[br][l], Bt[br][l], Cout, M);

            float* t = ping; ping = pong; pong = t;
        }

        // 4) max-pool over K into the output slab
        dim3 pgrid((NQ + 255) / 256, couts[br][2], BATCH);
        maxpool_kernel<<<pgrid, 256, 0, stream>>>(ping, out_feat, K, coffs[br]);
    }
}